// SimpleCNN_37323265802710
// MI455X (gfx1250) — compile-verified
//
#include <hip/hip_runtime.h>

typedef __attribute__((ext_vector_type(2))) float v2f;
typedef __attribute__((ext_vector_type(8))) float v8f;

#define NB    2048
#define C1    32
#define C2    64
#define HW    28
#define SPAT  784          // 28*28
#define HP    30           // padded edge
#define SPATP 900          // 30*30 padded spatial
#define K2    288          // 32*3*3
#define PSP   196          // 14*14 pooled spatial
#define FCK   12544        // 64*14*14
#define FCM   128

// ---------------------------------------------------------------------------
// DCLS v1 kernel construction: zero then scatter-add with hard rounding.
// ---------------------------------------------------------------------------
__global__ void dcls_zero(float* __restrict__ k1) {
    int i = blockIdx.x * 256 + threadIdx.x;
    if (i < C1 * 25) k1[i] = 0.0f;
}

__global__ void dcls_scatter(const float* __restrict__ w1,
                             const float* __restrict__ p1,
                             const float* __restrict__ p2,
                             float* __restrict__ k1) {
    int i = blockIdx.x * 256 + threadIdx.x;     // over 32*1*26
    if (i >= C1 * 26) return;
    float q1 = p1[i] + 2.0f;
    float q2 = p2[i] + 2.0f;
    float f1 = floorf(q1), f2 = floorf(q2);
    int j1 = (int)f1 + ((q1 - f1) >= 0.5f ? 1 : 0);   // straight-through round
    int j2 = (int)f2 + ((q2 - f2) >= 0.5f ? 1 : 0);
    if (j1 < 0 || j1 >= 5 || j2 < 0 || j2 >= 5) return;  // one_hot OOB -> drop
    int o = i / 26;
    atomicAdd(&k1[o * 25 + j1 * 5 + j2], w1[i]);
}

// ---------------------------------------------------------------------------
// conv1: (B,1,28,28) * (32,1,5,5) pad2 + bias + ReLU -> h1p (B,32,30,30),
// written WITH a zero halo so conv2's implicit-GEMM gathers need no bounds
// checks. Border lanes store 0 (idempotent across graph replays).
// ---------------------------------------------------------------------------
__global__ __launch_bounds__(256) void conv1_pad_kernel(
        const float* __restrict__ x, const float* __restrict__ k1,
        const float* __restrict__ b1, float* __restrict__ h1p) {
    __shared__ float sk[C1 * 25];
    for (int i = threadIdx.x; i < C1 * 25; i += 256) sk[i] = k1[i];
    __syncthreads();
    unsigned idx = blockIdx.x * 256u + threadIdx.x;   // (b,co,30,30) flat
    int s = idx % SPATP; unsigned t = idx / SPATP;
    int co = t % C1;     int b = t / C1;
    int yp = s / HP, xp = s % HP;

    float r = 0.0f;
    if (yp >= 1 && yp <= HW && xp >= 1 && xp <= HW) {
        int yh = yp - 1, xw = xp - 1;
        const float* xin = x + (long)b * SPAT;
        const float* wk  = sk + co * 25;
        float acc = b1[co];
        #pragma unroll
        for (int ty = 0; ty < 5; ++ty) {
            int yy = yh + ty - 2;
            if (yy < 0 || yy >= HW) continue;
            #pragma unroll
            for (int tx = 0; tx < 5; ++tx) {
                int xx = xw + tx - 2;
                if (xx < 0 || xx >= HW) continue;
                acc += wk[ty * 5 + tx] * xin[yy * HW + xx];
            }
        }
        r = fmaxf(acc, 0.0f);
    }
    h1p[idx] = r;
}

// ---------------------------------------------------------------------------
// conv2 + bias + ReLU + 2x2 maxpool, fused, on V_WMMA_F32_16X16X4_F32.
//   Implicit GEMM: M = Cout(64), K = Cin*9 = 288, N = pooled pos (B*196).
//   One wave owns ALL 4 m-tiles (64 couts) x 16 pooled outputs: per K-step
//   1 ds_load_b64 (offset table) + 2 activation gathers feed 4 WMMAs.
//   4 K-loop passes (2x2 sub-positions) with running max in registers;
//   zero halo in h1p makes every gather address valid -> no cndmask chains.
// ---------------------------------------------------------------------------
__global__ __launch_bounds__(256) void conv2_pool_wmma(
        const float* __restrict__ h1p, const float* __restrict__ w2,
        const float* __restrict__ b2, float* __restrict__ pooled) {
    __shared__ __align__(16) int tab[K2];   // cin*900 + dy*30 + dx
    for (int kk = threadIdx.x; kk < K2; kk += 256) {
        int cin = kk / 9, tp = kk % 9;
        int dy = tp / 3 - 1, dx = tp % 3 - 1;
        tab[kk] = cin * SPATP + dy * HP + dx;
    }
    __syncthreads();

    int wave = threadIdx.x >> 5;
    int lane = threadIdx.x & 31;
    int half = lane >> 4;
    int l15  = lane & 15;

    int nt = blockIdx.x * 8 + wave;          // 25088 n-tiles
    int pn = nt * 16 + l15;                  // pooled flat index b*196 + pi
    int b  = pn / PSP;
    int pi = pn - b * PSP;
    int py = pi / 14, px = pi - py * 14;

    const float* xin   = h1p + (long)b * C1 * SPATP;
    const float* wbase = w2 + l15 * K2;      // + mt*16*K2 per m-tile

    v8f best[4];
    #pragma unroll
    for (int mt = 0; mt < 4; ++mt)
        #pragma unroll
        for (int v = 0; v < 8; ++v) best[mt][v] = -3.402823466e38f;

    #pragma unroll
    for (int sub = 0; sub < 4; ++sub) {
        // padded coords: +1 halo offset
        int s = (py * 2 + (sub >> 1) + 1) * HP + (px * 2 + (sub & 1) + 1);
        v8f acc[4] = {};
        for (int k = 0; k < K2; k += 4) {
            int ka = k + half * 2;
            int2 off = *(const int2*)(tab + ka);       // ds_load_b64
            float bv0 = xin[s + off.x];                // always in-bounds
            float bv1 = xin[s + off.y];
            v2f bvec = {bv0, bv1};
            #pragma unroll
            for (int mt = 0; mt < 4; ++mt) {
                v2f a = *(const v2f*)(wbase + mt * 16 * K2 + ka);
                acc[mt] = __builtin_amdgcn_wmma_f32_16x16x4_f32(
                              false, a, false, bvec, (short)0, acc[mt],
                              false, false);
            }
        }
        #pragma unroll
        for (int mt = 0; mt < 4; ++mt)
            #pragma unroll
            for (int v = 0; v < 8; ++v)
                best[mt][v] = fmaxf(best[mt][v], acc[mt][v]);
    }

    // relu(max + bias): bias uniform over the 2x2 window, relu monotone.
    long base = (long)b * C2 * PSP + pi;
    #pragma unroll
    for (int mt = 0; mt < 4; ++mt) {
        #pragma unroll
        for (int v = 0; v < 8; ++v) {
            int mo = mt * 16 + v + half * 8;
            pooled[base + (long)mo * PSP] = fmaxf(best[mt][v] + b2[mo], 0.0f);
        }
    }
}

// ---------------------------------------------------------------------------
// fc1 GEMM on V_WMMA_F32_16X16X4_F32: M=128 outputs, N=2048 batch, K=12544.
// Both A (weight row) and B (activation row) K-pairs are contiguous -> b64.
// Streaming K-loop prefetches 1KB ahead via global_prefetch_b8.
// ---------------------------------------------------------------------------
__global__ __launch_bounds__(256) void fc1_wmma(
        const float* __restrict__ pooled, const float* __restrict__ fc1w,
        const float* __restrict__ fc1b, float* __restrict__ fc1o) {
    int wave = threadIdx.x >> 5;
    int lane = threadIdx.x & 31;
    int half = lane >> 4;
    int l15  = lane & 15;

    int tile = blockIdx.x * 8 + wave;     // 8 m-tiles * 128 n-tiles = 1024
    int mt = tile & 7;
    int nt = tile >> 3;

    const float* wrow = fc1w   + (long)(mt * 16 + l15) * FCK;
    const float* hrow = pooled + (long)(nt * 16 + l15) * FCK;

    v8f acc = {};
    for (int kb = 0; kb < FCK; kb += 256) {
        __builtin_prefetch(wrow + kb + 256, 0, 0);   // global_prefetch_b8
        __builtin_prefetch(hrow + kb + 256, 0, 0);
        for (int k = kb; k < kb + 256; k += 4) {
            int ka = k + half * 2;
            v2f a  = *(const v2f*)(wrow + ka);
            v2f bv = *(const v2f*)(hrow + ka);
            acc = __builtin_amdgcn_wmma_f32_16x16x4_f32(
                      false, a, false, bv, (short)0, acc, false, false);
        }
    }

    int n = nt * 16 + l15;
    #pragma unroll
    for (int v = 0; v < 8; ++v) {
        int mo = mt * 16 + v + half * 8;
        fc1o[(long)n * FCM + mo] = fmaxf(acc[v] + fc1b[mo], 0.0f);
    }
}

// ---------------------------------------------------------------------------
// fc2: (2048,128) @ (10,128)^T + bias  -> d_out (2048,10). Tiny.
// ---------------------------------------------------------------------------
__global__ void fc2_kernel(const float* __restrict__ fc1o,
                           const float* __restrict__ fc2w,
                           const float* __restrict__ fc2b,
                           float* __restrict__ out) {
    int idx = blockIdx.x * 256 + threadIdx.x;
    if (idx >= NB * 10) return;
    int j = idx % 10, b = idx / 10;
    const float* h = fc1o + (long)b * FCM;
    const float* w = fc2w + j * FCM;
    float acc = fc2b[j];
    #pragma unroll 8
    for (int c = 0; c < FCM; ++c) acc += h[c] * w[c];
    out[idx] = acc;
}

// ---------------------------------------------------------------------------
extern "C" void kernel_launch(void* const* d_in, const int* in_sizes, int n_in,
                              void* d_out, int out_size, void* d_ws, size_t ws_size,
                              hipStream_t stream) {
    const float* x     = (const float*)d_in[0];
    const float* w1    = (const float*)d_in[1];
    const float* p1    = (const float*)d_in[2];
    const float* p2    = (const float*)d_in[3];
    const float* b1    = (const float*)d_in[4];
    const float* w2    = (const float*)d_in[5];
    const float* b2    = (const float*)d_in[6];
    const float* fc1w  = (const float*)d_in[7];
    const float* fc1b  = (const float*)d_in[8];
    const float* fc2w  = (const float*)d_in[9];
    const float* fc2b  = (const float*)d_in[10];
    float* out = (float*)d_out;

    // workspace layout (bytes):
    //   [0, 4096)                 : k1 (32*25 f32)
    //   [4096, +235929600)        : h1p    (B,32,30,30)  zero-halo, 236 MB
    //   [+h1p, +102760448)        : pooled (B,64,14,14)  102.8 MB
    //   [+pooled, +1048576)       : fc1 output (B,128)     1.0 MB
    char* ws = (char*)d_ws;
    float* k1     = (float*)(ws);
    float* h1p    = (float*)(ws + 4096);
    float* pooled = (float*)(ws + 4096 + 235929600ull);
    float* fc1o   = (float*)(ws + 4096 + 235929600ull + 102760448ull);

    dcls_zero   <<<4, 256, 0, stream>>>(k1);
    dcls_scatter<<<4, 256, 0, stream>>>(w1, p1, p2, k1);

    // conv1: 2048*32*900 = 58,982,400 threads / 256 = 230,400 blocks (exact)
    conv1_pad_kernel<<<230400, 256, 0, stream>>>(x, k1, b1, h1p);

    // fused conv2+pool: 2048*196/16 = 25,088 wave-tiles / 8 = 3,136 blocks
    conv2_pool_wmma<<<3136, 256, 0, stream>>>(h1p, w2, b2, pooled);

    // fc1: (128/16)*(2048/16) = 1024 wave-tiles / 8 = 128 blocks
    fc1_wmma<<<128, 256, 0, stream>>>(pooled, fc1w, fc1b, fc1o);

    // fc2: 20480 outputs
    fc2_kernel<<<80, 256, 0, stream>>>(fc1o, fc2w, fc2b, out);
}